// HybridQuantumKernelRegressor_65481071396126
// MI455X (gfx1250) — compile-verified
//
#include <hip/hip_runtime.h>
#include <math.h>

// ---------------------------------------------------------------------------
// Hybrid quantum-kernel ridge regression for MI455X (gfx1250, wave32).
//   K[i][j]   = exp(-||xi-xj||^2) * |prod_d cos(0.5*(xi_d - xj_d))|
//   alpha     = (K + reg I)^-1 y      (fixed-iteration CG; K is SPD, L2-resident)
//   y_pred    = K_test @ alpha        (K_test stored transposed -> same WMMA matvec)
// Matvec hot loop: V_WMMA_F32_16X16X4_F32, p broadcast into all A rows,
// coalesced row-wise B tiles with compile-time row stride (immediate offsets),
// unconditional global_prefetch_b8 stream-ahead, dual accumulator chains.
// ---------------------------------------------------------------------------

#define N_DIM        4096
#define REG          1e-3f
#define CG_ITERS     512
#define KCHUNK       512
#define NCHUNKS      (N_DIM / KCHUNK)     // 8 split-K chunks
#define RED_BLOCKS   64
#define RED_THREADS  256
#define MV_WAVES_PER_BLOCK 8
#define MV_BLOCK     (MV_WAVES_PER_BLOCK * 32)

typedef __attribute__((ext_vector_type(2))) float v2f;
typedef __attribute__((ext_vector_type(8))) float v8f;

// --------------------------- kernel-matrix build ---------------------------

__device__ __forceinline__ float hybrid_val(float4 a0, float4 a1,
                                            float4 b0, float4 b1) {
  float d0 = a0.x - b0.x, d1 = a0.y - b0.y, d2 = a0.z - b0.z, d3 = a0.w - b0.w;
  float d4 = a1.x - b1.x, d5 = a1.y - b1.y, d6 = a1.z - b1.z, d7 = a1.w - b1.w;
  float sq = d0*d0 + d1*d1 + d2*d2 + d3*d3 + d4*d4 + d5*d5 + d6*d6 + d7*d7;
  float cp = __cosf(0.5f*d0) * __cosf(0.5f*d1) * __cosf(0.5f*d2) * __cosf(0.5f*d3)
           * __cosf(0.5f*d4) * __cosf(0.5f*d5) * __cosf(0.5f*d6) * __cosf(0.5f*d7);
  return __expf(-sq) * fabsf(cp);          // GAMMA == 1.0
}

// K[i][j] over X_train x X_train, + reg on the diagonal.
__global__ void build_K_train(const float* __restrict__ X, float* __restrict__ K) {
  int j = blockIdx.x * 16 + threadIdx.x;   // column (coalesced store)
  int i = blockIdx.y * 16 + threadIdx.y;   // row
  const float4* X4 = (const float4*)X;     // 8 floats per row = 2 x float4
  float4 xi0 = X4[2*i], xi1 = X4[2*i + 1];
  float4 xj0 = X4[2*j], xj1 = X4[2*j + 1];
  float v = hybrid_val(xi0, xi1, xj0, xj1);
  if (i == j) v += REG;
  K[(size_t)i * N_DIM + j] = v;
}

// KT[j_train][i_test] = k(x_test_i, x_train_j): test kernel stored TRANSPOSED
// so the WMMA matvec can read its rows coalesced.
__global__ void build_KT_test(const float* __restrict__ Xte,
                              const float* __restrict__ Xtr,
                              float* __restrict__ KT) {
  int i = blockIdx.x * 16 + threadIdx.x;   // test index (column, coalesced)
  int j = blockIdx.y * 16 + threadIdx.y;   // train index (row)
  const float4* A4 = (const float4*)Xte;
  const float4* B4 = (const float4*)Xtr;
  float4 xi0 = A4[2*i], xi1 = A4[2*i + 1];
  float4 xj0 = B4[2*j], xj1 = B4[2*j + 1];
  KT[(size_t)j * N_DIM + i] = hybrid_val(xi0, xi1, xj0, xj1);
}

// --------------------------- WMMA split-K matvec ---------------------------
// y[n] = sum_k A[k][n] * p[k]   (A row-major; rows are the K dimension)
// Per wave: one 16-wide n-tile, one 512-long k-chunk.
//   A-frag (16x4 f32): all 16 M-rows = p[k..k+3]; VGPR0 holds K={0|2},
//     VGPR1 holds K={1|3}, selected by half-wave (ISA 7.12.2 layout).
//   B-frag (4x16 f32): B[k][n] = A[k][n0+lane%16] -> coalesced row reads with
//     compile-time row stride N_DIM (folds into global_load immediate offsets).
//   Every row of C equals the y tile; lane<16 stores C VGPR0.
__global__ void matvec_wmma_partial(const float* __restrict__ A,
                                    const float* __restrict__ p,
                                    float* __restrict__ part) { // [NCHUNKS][N]
  const int wave  = (int)(blockIdx.x * (blockDim.x >> 5) + (threadIdx.x >> 5));
  const int lane  = threadIdx.x & 31;
  const int ntile = wave & ((N_DIM >> 4) - 1);
  const int chunk = wave >> (12 - 4);            // wave / (N_DIM/16)
  const int n0    = ntile << 4;
  const int half  = lane >> 4;                   // 0: lanes 0-15, 1: lanes 16-31
  const int col   = n0 + (lane & 15);
  const int k0    = chunk * KCHUNK;

  const size_t R = (size_t)N_DIM;                // compile-time row stride
  const float* bp = A + (size_t)(k0 + (half << 1)) * R + col;
  const float* pp = p + k0 + (half << 1);

  v8f acc0 = {}, acc1 = {};
  #pragma unroll 1
  for (int it = 0; it < KCHUNK / 16; ++it) {
    // speculative stream-ahead of the next 16-row block (global_prefetch_b8)
    __builtin_prefetch(bp + 16 * R, 0, 3);
    v2f a0, a1, a2, a3, b0, b1, b2, b3;
    a0.x = pp[0];  a0.y = pp[1];
    a1.x = pp[4];  a1.y = pp[5];
    a2.x = pp[8];  a2.y = pp[9];
    a3.x = pp[12]; a3.y = pp[13];
    b0.x = bp[0];      b0.y = bp[R];
    b1.x = bp[4 * R];  b1.y = bp[5 * R];
    b2.x = bp[8 * R];  b2.y = bp[9 * R];
    b3.x = bp[12 * R]; b3.y = bp[13 * R];
    acc0 = __builtin_amdgcn_wmma_f32_16x16x4_f32(false, a0, false, b0,
                                                 (short)0, acc0, false, false);
    acc1 = __builtin_amdgcn_wmma_f32_16x16x4_f32(false, a1, false, b1,
                                                 (short)0, acc1, false, false);
    acc0 = __builtin_amdgcn_wmma_f32_16x16x4_f32(false, a2, false, b2,
                                                 (short)0, acc0, false, false);
    acc1 = __builtin_amdgcn_wmma_f32_16x16x4_f32(false, a3, false, b3,
                                                 (short)0, acc1, false, false);
    bp += 16 * R;
    pp += 16;
  }
  // C layout: VGPR0, lanes 0-15 -> (M=0, N=lane); all M rows identical.
  if (lane < 16) part[(size_t)chunk * N_DIM + col] = acc0[0] + acc1[0];
}

__global__ void reduce_partials(const float* __restrict__ part,
                                float* __restrict__ y) {
  int i = blockIdx.x * blockDim.x + threadIdx.x;
  float s = 0.f;
  #pragma unroll
  for (int c = 0; c < NCHUNKS; ++c) s += part[(size_t)c * N_DIM + i];
  y[i] = s;
}

// ------------------------------ CG helpers --------------------------------

__global__ void cg_init(const float* __restrict__ b, float* __restrict__ x,
                        float* __restrict__ r, float* __restrict__ p) {
  int i = blockIdx.x * blockDim.x + threadIdx.x;
  x[i] = 0.f; float bi = b[i]; r[i] = bi; p[i] = bi;
}

__global__ void dot_partial(const float* __restrict__ a,
                            const float* __restrict__ b,
                            float* __restrict__ out, int n) {
  __shared__ float sm[RED_THREADS];
  float s = 0.f;
  for (int i = blockIdx.x * blockDim.x + threadIdx.x; i < n;
       i += gridDim.x * blockDim.x)
    s += a[i] * b[i];
  sm[threadIdx.x] = s;
  __syncthreads();
  for (int off = RED_THREADS >> 1; off > 0; off >>= 1) {
    if ((int)threadIdx.x < off) sm[threadIdx.x] += sm[threadIdx.x + off];
    __syncthreads();
  }
  if (threadIdx.x == 0) out[blockIdx.x] = sm[0];
}

// scal: [0]=rr_old  [1]=alpha  [2]=beta
__global__ void finalize_rr0(const float* __restrict__ part, float* scal) {
  float s = 0.f;
  for (int i = 0; i < RED_BLOCKS; ++i) s += part[i];
  scal[0] = s;
}

__global__ void finalize_alpha(const float* __restrict__ part, float* scal) {
  float s = 0.f;                           // s = p.Ap
  for (int i = 0; i < RED_BLOCKS; ++i) s += part[i];
  scal[1] = scal[0] / s;
}

__global__ void update_xr(float* __restrict__ x, float* __restrict__ r,
                          const float* __restrict__ p, const float* __restrict__ v,
                          const float* __restrict__ scal,
                          float* __restrict__ part, int n) {
  __shared__ float sm[RED_THREADS];
  float alpha = scal[1];
  float s = 0.f;
  for (int i = blockIdx.x * blockDim.x + threadIdx.x; i < n;
       i += gridDim.x * blockDim.x) {
    x[i] += alpha * p[i];
    float ri = r[i] - alpha * v[i];
    r[i] = ri;
    s += ri * ri;
  }
  sm[threadIdx.x] = s;
  __syncthreads();
  for (int off = RED_THREADS >> 1; off > 0; off >>= 1) {
    if ((int)threadIdx.x < off) sm[threadIdx.x] += sm[threadIdx.x + off];
    __syncthreads();
  }
  if (threadIdx.x == 0) part[blockIdx.x] = sm[0];
}

__global__ void finalize_beta(const float* __restrict__ part, float* scal) {
  float s = 0.f;                           // s = rr_new
  for (int i = 0; i < RED_BLOCKS; ++i) s += part[i];
  scal[2] = s / scal[0];
  scal[0] = s;
}

__global__ void update_p(float* __restrict__ p, const float* __restrict__ r,
                         const float* __restrict__ scal) {
  int i = blockIdx.x * blockDim.x + threadIdx.x;
  p[i] = r[i] + scal[2] * p[i];
}

// ------------------------------- launcher ----------------------------------

extern "C" void kernel_launch(void* const* d_in, const int* in_sizes, int n_in,
                              void* d_out, int out_size, void* d_ws, size_t ws_size,
                              hipStream_t stream) {
  const int n = N_DIM;
  const float* X_train = (const float*)d_in[0];   // [4096, 8] f32
  const float* y_train = (const float*)d_in[1];   // [4096]    f32
  const float* X_test  = (const float*)d_in[2];   // [4096, 8] f32
  float* out = (float*)d_out;                     // [4096]    f32

  // workspace carve-up (floats)
  float* wsf   = (float*)d_ws;
  float* K     = wsf;                                  // 64 MB (L2-resident)
  float* KT    = K  + (size_t)n * n;                   // 64 MB (L2-resident)
  float* x     = KT + (size_t)n * n;
  float* r     = x  + n;
  float* p     = r  + n;
  float* v     = p  + n;
  float* pmv   = v  + n;                               // [NCHUNKS][n]
  float* pred  = pmv + (size_t)NCHUNKS * n;
  float* scal  = pred + RED_BLOCKS;                    // [rr_old, alpha, beta]

  dim3 b16(16, 16);
  dim3 gK(n / 16, n / 16);
  const int mv_blocks  = ((n / 16) * NCHUNKS) / MV_WAVES_PER_BLOCK;  // 256
  const int vec_blocks = n / 256;

  // 1) Build both kernel matrices (K_test stored transposed).
  build_K_train<<<gK, b16, 0, stream>>>(X_train, K);
  build_KT_test<<<gK, b16, 0, stream>>>(X_test, X_train, KT);

  // 2) CG init: x=0, r=p=y, rr_old = r.r
  cg_init<<<vec_blocks, 256, 0, stream>>>(y_train, x, r, p);
  dot_partial<<<RED_BLOCKS, RED_THREADS, 0, stream>>>(r, r, pred, n);
  finalize_rr0<<<1, 1, 0, stream>>>(pred, scal);

  // 3) Fixed-iteration CG (deterministic; all reductions fixed-order).
  for (int it = 0; it < CG_ITERS; ++it) {
    matvec_wmma_partial<<<mv_blocks, MV_BLOCK, 0, stream>>>(K, p, pmv);
    reduce_partials<<<vec_blocks, 256, 0, stream>>>(pmv, v);
    dot_partial<<<RED_BLOCKS, RED_THREADS, 0, stream>>>(p, v, pred, n);
    finalize_alpha<<<1, 1, 0, stream>>>(pred, scal);
    update_xr<<<RED_BLOCKS, RED_THREADS, 0, stream>>>(x, r, p, v, scal, pred, n);
    finalize_beta<<<1, 1, 0, stream>>>(pred, scal);
    update_p<<<vec_blocks, 256, 0, stream>>>(p, r, scal);
  }

  // 4) Predict: y_pred = K_test @ alpha == (KT rows) . x  -> same WMMA matvec.
  matvec_wmma_partial<<<mv_blocks, MV_BLOCK, 0, stream>>>(KT, x, pmv);
  reduce_partials<<<vec_blocks, 256, 0, stream>>>(pmv, out);
}